// LorentzFrechetMean_25615184953684
// MI455X (gfx1250) — compile-verified
//
#include <hip/hip_runtime.h>
#include <hip/hip_bf16.h>

// LorentzFrechetMean on MI455X (gfx1250, wave32).
// One workgroup per batch (512 blocks x 256 threads = 8 waves).
// points (128MB) fits in the 192MB L2 -> 15 iterations are L2-resident.
// Inner products via V_WMMA_F32_16X16X4_F32 (B-matrix = ytilde broadcast
// across all 16 columns, so every C column holds the 16 inners).
// Tile staging via GLOBAL_LOAD_ASYNC_TO_LDS_B128 (ASYNCcnt, no VGPR round trip).

typedef float v2f __attribute__((ext_vector_type(2)));
typedef float v8f __attribute__((ext_vector_type(8)));

#define N_PTS   1024
#define DIM     64
#define WAVES   8
#define TILE    16
#define TSTRIDE 68   // 64 + 4 pad: WMMA-A reads and dim-major reads bank-conflict-free

__device__ __forceinline__ float waveSum32(float v) {
    v += __shfl_xor(v, 1, 32);
    v += __shfl_xor(v, 2, 32);
    v += __shfl_xor(v, 4, 32);
    v += __shfl_xor(v, 8, 32);
    v += __shfl_xor(v, 16, 32);
    return v;
}

// stable arccosh(1+u), u pre-clipped >= 1e-7
__device__ __forceinline__ float acosh1p(float u) {
    return log1pf(u + sqrtf(u * (u + 2.0f)));
}

// branchless select that clang cannot canonicalize into a dynamic vector
// extract (which would expand to a cmp/cndmask tree): integer bit-blend.
__device__ __forceinline__ float bitSelect(unsigned mask, float e1, float e0) {
    return __uint_as_float((__float_as_uint(e1) & mask) |
                           (__float_as_uint(e0) & ~mask));
}

// async 16B global -> LDS copy (per lane); generic LDS pointer's low 32 bits
// are the wave-relative LDS byte offset (flat aperture truncation).
__device__ __forceinline__ void asyncCopy16(const float* gp, float* lp) {
#if defined(__gfx1250__)
    unsigned lo = (unsigned)(uintptr_t)lp;
    asm volatile("global_load_async_to_lds_b128 %0, %1, off"
                 :: "v"(lo), "v"((unsigned long long)(uintptr_t)gp)
                 : "memory");
#else
    *(float4*)lp = *(const float4*)gp;
#endif
}

__device__ __forceinline__ void asyncWaitAll() {
#if defined(__gfx1250__)
    asm volatile("s_wait_asynccnt 0x0" ::: "memory");
#endif
}

__global__ __launch_bounds__(256)
void frechet_kernel(const float* __restrict__ points,
                    const float* __restrict__ weights,
                    float* __restrict__ out)
{
    __shared__ __align__(16) float sh_tile[WAVES][TILE][TSTRIDE];
    __shared__ float sh_wnorm[N_PTS];
    __shared__ float sh_y[DIM];      // current estimate y
    __shared__ float sh_yt[DIM];     // ytilde = y with time coord negated
    __shared__ float sh_wacc[WAVES][DIM];
    __shared__ float sh_wcs[WAVES];
    __shared__ float sh_cbuf[WAVES][TILE];
    __shared__ float sh_red[256];
    __shared__ float sh_rv[256];
    __shared__ int   sh_ri[256];
    __shared__ int   sh_done;

    const int tid  = threadIdx.x;
    const int lane = tid & 31;
    const int wv   = tid >> 5;
    const int hi   = lane >> 4;          // half-wave select (K offset +2 / M offset +8)
    const int rrow = lane & 15;          // A-matrix row owned by this lane
    const int par  = lane & 1;           // parity: which of each acc pair we own
    const unsigned pmask = (unsigned)(-par);   // 0 or 0xFFFFFFFF
    const int b    = blockIdx.x;

    const float* P = points  + (size_t)b * N_PTS * DIM;
    const float* W = weights + (size_t)b * N_PTS;

    // ---------------- init: weight sum + argmax (y0 = point with max weight) ------
    float lsum = 0.f, bvv = -1e30f;
    int bii = 0;
    for (int i = tid; i < N_PTS; i += 256) {
        float w = W[i];
        lsum += w;
        if (w > bvv) { bvv = w; bii = i; }   // ascending -> first occurrence on ties
    }
    sh_red[tid] = lsum; sh_rv[tid] = bvv; sh_ri[tid] = bii;
    __syncthreads();
    for (int s = 128; s > 0; s >>= 1) {
        if (tid < s) {
            sh_red[tid] += sh_red[tid + s];
            float ov = sh_rv[tid + s]; int oi = sh_ri[tid + s];
            if (ov > sh_rv[tid] || (ov == sh_rv[tid] && oi < sh_ri[tid])) {
                sh_rv[tid] = ov; sh_ri[tid] = oi;
            }
        }
        __syncthreads();
    }
    const float invsum = 1.0f / sh_red[0];
    const int   idx0   = sh_ri[0];
    for (int i = tid; i < N_PTS; i += 256) sh_wnorm[i] = W[i] * invsum;
    if (tid < DIM) {
        float yv = P[(size_t)idx0 * DIM + tid];
        sh_y[tid]  = yv;
        sh_yt[tid] = (tid == 0) ? -yv : yv;
    }
    if (tid == 0) sh_done = 0;
    __syncthreads();

    // ---------------- fixed-point iterations ------------------------------------
    for (int it = 0; it < 15; ++it) {
        float a0 = 0.f, a1 = 0.f;     // wt partial for dims 2*lane, 2*lane+1
        float csum = 0.f;             // per-(half,parity) partial of sum(c_i*inner_i)

        for (int t = 0; t < 8; ++t) {
            const int i0 = wv * 128 + t * TILE;

            // ---- stage 16x64 tile: 8 async b128 copies, all in flight ----------
            #pragma unroll
            for (int r = 0; r < 8; ++r) {
                int lin = r * 32 + lane;
                int row = lin >> 4, c4 = lin & 15;
                asyncCopy16(P + (size_t)(i0 + row) * DIM + (c4 << 2),
                            &sh_tile[wv][row][c4 << 2]);
                // warm L2 for the next tile; wrap keeps address in-batch
                int nrow = (i0 + TILE + row) & (N_PTS - 1);
                __builtin_prefetch((const void*)(P + (size_t)nrow * DIM + (c4 << 2)), 0, 1);
            }
            asyncWaitAll();   // ASYNCcnt -> 0: tile resident in LDS

            // ---- phase 1: inner_i = <p_i, ytilde> via f32 WMMA 16x16x4 ----------
            v8f accA = {0.f,0.f,0.f,0.f,0.f,0.f,0.f,0.f};
            v8f accB = {0.f,0.f,0.f,0.f,0.f,0.f,0.f,0.f};
#if defined(__gfx1250__) && __has_builtin(__builtin_amdgcn_wmma_f32_16x16x4_f32)
            #pragma unroll
            for (int k0 = 0; k0 < DIM; k0 += 8) {
                const int colA = k0 + 2 * hi;          // even K chunk
                const int colB = k0 + 4 + 2 * hi;      // odd  K chunk
                v2f aA = *(const v2f*)&sh_tile[wv][rrow][colA];
                v2f bA = *(const v2f*)&sh_yt[colA];
                v2f aB = *(const v2f*)&sh_tile[wv][rrow][colB];
                v2f bB = *(const v2f*)&sh_yt[colB];
                accA = __builtin_amdgcn_wmma_f32_16x16x4_f32(
                         false, aA, false, bA, (short)0, accA, false, false);
                accB = __builtin_amdgcn_wmma_f32_16x16x4_f32(
                         false, aB, false, bB, (short)0, accB, false, false);
            }
#else
            #pragma unroll
            for (int r = 0; r < 8; ++r) {
                float s = 0.f;
                for (int k = 0; k < DIM; ++k)
                    s += sh_tile[wv][hi * 8 + r][k] * sh_yt[k];
                accA[r] = s; accB[r] = 0.f;
            }
#endif
            v8f acc = accA + accB;

            // acc[r] = inner_{i0 + hi*8 + r}, replicated across the 16 N-lanes.
            // Parity split: even lanes process acc[0,2,4,6], odd lanes acc[1,3,5,7]
            // -> 4 transcendental chains per lane instead of 8. Selection done with
            // an integer bit-blend so it stays O(1) VALU (no dynamic-extract tree).
            #pragma unroll
            for (int k = 0; k < 4; ++k) {
                float inner = bitSelect(pmask, acc[2 * k + 1], acc[2 * k]);
                int   ip    = i0 + hi * 8 + 2 * k + par;
                float u  = fmaxf(-inner - 1.0f, 1e-7f);    // clip(-inner,1+eps) - 1
                float d  = acosh1p(u);                      // geodesic dist
                float sd = fmaxf(d, 1e-8f);
                float e  = __expf(sd);
                float sh = 0.5f * (e - 1.0f / e);           // sinh(sd), sd >= 0
                float scale = (fabsf(sh) > 1e-8f) ? (sd / sh) : 1.0f;
                float ci = sh_wnorm[ip] * scale;
                csum += ci * inner;
                // 8 lanes hold the identical ci: same-address LDS store, one lane
                // wins, value identical -> branch-free broadcast
                sh_cbuf[wv][hi * 8 + 2 * k + par] = ci;
            }

            // ---- phase 2: wt_partial += c_i * p_i (lane owns dims 2L, 2L+1) ----
            #pragma unroll
            for (int i = 0; i < TILE; ++i) {
                float ci = sh_cbuf[wv][i];                  // broadcast read
                v2f  p2  = *(const v2f*)&sh_tile[wv][i][2 * lane];
                a0 = fmaf(ci, p2[0], a0);
                a1 = fmaf(ci, p2[1], a1);
            }
        }

        // ---- cross-wave reduction staging ----
        float cs1   = csum + __shfl_xor(csum, 1, 32);       // merge parities
        float cstot = cs1  + __shfl_xor(cs1, 16, 32);       // merge halves
        v2f av; av[0] = a0; av[1] = a1;
        *(v2f*)&sh_wacc[wv][2 * lane] = av;
        if (lane == 0) sh_wcs[wv] = cstot;
        __syncthreads();

        // ---- wave 0: finish reduction, update y (32 lanes x 2 dims = 64) ----
        if (wv == 0) {
            float sx = 0.f, sy = 0.f, CS = 0.f;
            #pragma unroll
            for (int w = 0; w < WAVES; ++w) {
                v2f wa = *(const v2f*)&sh_wacc[w][2 * lane];
                sx += wa[0]; sy += wa[1];
                CS += sh_wcs[w];
            }
            float ya = sh_y[2 * lane], yb = sh_y[2 * lane + 1];
            float wtx = sx - CS * ya;                        // wt = sum c p - (sum c inner) y
            float wty = sy - CS * yb;

            float tot = waveSum32(wtx * wtx + wty * wty);
            float wt0 = __shfl(wtx, 0, 32);
            float msq = tot - 2.0f * wt0 * wt0;              // mink(wt,wt)
            float v   = sqrtf(fmaxf(-msq, 1e-11f));          // safe_sqrt(-mink)

            float ev  = __expf(v);
            float iev = 1.0f / ev;
            float chv = 0.5f * (ev + iev);
            float svv = 0.5f * (ev - iev) / v;               // sinh(v)/v
            float rx = chv * ya + svv * wtx;
            float ry = chv * yb + svv * wty;

            // project_to_manifold: recompute time coordinate
            float spt = waveSum32(rx * rx + ry * ry);
            float r0  = __shfl(rx, 0, 32);
            float t0  = sqrtf(1.0f + (spt - r0 * r0));
            float nx  = (lane == 0) ? t0 : rx;
            float ny  = ry;
            if (v < 1e-8f) { nx = ya; ny = yb; }             // keep y if degenerate

            // convergence metric (cannot go below TOL due to 1+1e-7 clip; fidelity)
            float ipt = waveSum32(nx * ya + ny * yb);
            float p00 = __shfl(nx * ya, 0, 32);
            float mk  = ipt - 2.0f * p00;                    // mink(y_new, y)
            float uu  = fmaxf(-mk - 1.0f, 1e-7f);
            float change = acosh1p(uu);

            sh_y[2 * lane]      = nx;
            sh_y[2 * lane + 1]  = ny;
            sh_yt[2 * lane]     = (lane == 0) ? -nx : nx;
            sh_yt[2 * lane + 1] = ny;
            if (lane == 0) sh_done = (change < 1e-7f) ? 1 : 0;
        }
        __syncthreads();
        if (sh_done) break;
    }

    if (tid < DIM) out[(size_t)b * DIM + tid] = sh_y[tid];
}

extern "C" void kernel_launch(void* const* d_in, const int* in_sizes, int n_in,
                              void* d_out, int out_size, void* d_ws, size_t ws_size,
                              hipStream_t stream) {
    const float* points  = (const float*)d_in[0];
    const float* weights = (const float*)d_in[1];
    float* out = (float*)d_out;
    const int B = in_sizes[1] / N_PTS;   // 512
    frechet_kernel<<<B, 256, 0, stream>>>(points, weights, out);
}